// _GINRegressionNet_48137993453578
// MI455X (gfx1250) — compile-verified
//
#include <hip/hip_runtime.h>
#include <hip/hip_bf16.h>

#define N_NODES 100000
#define N_EDGES 1600000
#define DIM 64
#define LAYERS 4
#define N_GRAPHS 512
#define BN_EPS 1e-5f

typedef float v2f __attribute__((ext_vector_type(2)));
typedef float v8f __attribute__((ext_vector_type(8)));

// agg[i] = (1+eps[layer]) * h[i]
__global__ void k_init_agg(const float* __restrict__ h, float* __restrict__ agg,
                           const float* __restrict__ eps, int layer) {
    int i = blockIdx.x * blockDim.x + threadIdx.x;
    if (i < N_NODES * DIM) agg[i] = (1.0f + eps[layer]) * h[i];
}

// One wave (32 lanes) per edge; each lane handles 2 columns -> coalesced 256B gather,
// f32 atomic scatter (addresses spread over 100k*64 slots -> near-zero contention, L2 resident).
__global__ void k_scatter(const float* __restrict__ h, const int* __restrict__ src,
                          const int* __restrict__ dst, float* __restrict__ agg) {
    int t = blockIdx.x * blockDim.x + threadIdx.x;
    int e = t >> 5;
    if (e >= N_EDGES) return;
    int col = (t & 31) * 2;
    int s = src[e], d = dst[e];
    float2 v = *(const float2*)(h + (size_t)s * DIM + col);
    atomicAdd(agg + (size_t)d * DIM + col + 0, v.x);
    atomicAdd(agg + (size_t)d * DIM + col + 1, v.y);
}

// Z = A @ W + bias, fused per-column sum / sum-of-squares for BN.
// Block = 128 threads = 4 waves; wave w computes the 16x16 tile (rows blockIdx*16, cols w*16)
// via 16x V_WMMA_F32_16X16X4_F32 (K=64).
__global__ void __launch_bounds__(128)
k_gemm_stats(const float* __restrict__ A, const float* __restrict__ W,
             const float* __restrict__ bias, float* __restrict__ Z,
             float* __restrict__ colsum, float* __restrict__ colsq) {
    __shared__ float ssum[DIM];
    __shared__ float ssq[DIM];
    int tid = threadIdx.x;
    if (tid < DIM) { ssum[tid] = 0.0f; ssq[tid] = 0.0f; }
    __syncthreads();

    int wave = tid >> 5;          // 0..3 -> column tile
    int lane = tid & 31;
    int half = lane >> 4;         // k-group / row-half selector
    int idx  = lane & 15;
    int rowBase = blockIdx.x * 16;
    int col = wave * 16 + idx;

    v8f c = {0.f, 0.f, 0.f, 0.f, 0.f, 0.f, 0.f, 0.f};
    const float* Arow = A + (size_t)(rowBase + idx) * DIM;
    #pragma unroll
    for (int kt = 0; kt < DIM; kt += 4) {
        int kb = kt + half * 2;
        v2f a, b;
        a.x = Arow[kb];
        a.y = Arow[kb + 1];
        b.x = W[(size_t)kb * DIM + col];
        b.y = W[(size_t)(kb + 1) * DIM + col];
        c = __builtin_amdgcn_wmma_f32_16x16x4_f32(false, a, false, b,
                                                  (short)0, c, false, false);
    }

    float bv = bias[col];
    float lsum = 0.0f, lsq = 0.0f;
    #pragma unroll
    for (int v = 0; v < 8; ++v) {
        int row = rowBase + half * 8 + v;   // C layout: VGPR v -> M = v (+8 for lanes 16..31)
        float z = c[v] + bv;
        Z[(size_t)row * DIM + col] = z;
        lsum += z;
        lsq  += z * z;
    }
    atomicAdd(&ssum[col], lsum);            // ds_add_f32, 2 lanes per column
    atomicAdd(&ssq[col], lsq);
    __syncthreads();
    if (tid < DIM) {
        atomicAdd(colsum + tid, ssum[tid]); // 64 global atomics per block
        atomicAdd(colsq  + tid, ssq[tid]);
    }
}

// Fold BN into per-column scale/shift: g*(h-mu)*rsqrt(var+eps)+b = h*scale + shift
__global__ void k_bn_params(const float* __restrict__ colsum, const float* __restrict__ colsq,
                            const float* __restrict__ g, const float* __restrict__ bt,
                            float* __restrict__ scale, float* __restrict__ shift) {
    int c = threadIdx.x;
    if (c < DIM) {
        const float invN = 1.0f / (float)N_NODES;
        float mean = colsum[c] * invN;
        float var  = colsq[c] * invN - mean * mean;  // biased variance
        float sc = g[c] * rsqrtf(var + BN_EPS);
        scale[c] = sc;
        shift[c] = bt[c] - mean * sc;
    }
}

__global__ void k_bnrelu(const float* __restrict__ Z, const float* __restrict__ scale,
                         const float* __restrict__ shift, float* __restrict__ out) {
    int i = blockIdx.x * blockDim.x + threadIdx.x;
    if (i < N_NODES * DIM) {
        int c = i & (DIM - 1);
        float v = Z[i] * scale[c] + shift[c];
        out[i] = v > 0.0f ? v : 0.0f;
    }
}

__global__ void k_zero(float* __restrict__ p, int n) {
    int i = blockIdx.x * blockDim.x + threadIdx.x;
    if (i < n) p[i] = 0.0f;
}

__global__ void k_pool(const float* __restrict__ h, const int* __restrict__ batch,
                       float* __restrict__ pooled, float* __restrict__ counts) {
    int i = blockIdx.x * blockDim.x + threadIdx.x;
    if (i >= N_NODES * DIM) return;
    int node = i >> 6;
    int col  = i & 63;
    int g = batch[node];
    atomicAdd(pooled + (size_t)g * DIM + col, h[i]);
    if (col == 0) atomicAdd(counts + g, 1.0f);
}

__global__ void k_head(const float* __restrict__ pooled, const float* __restrict__ counts,
                       const float* __restrict__ Wh, const float* __restrict__ bh,
                       float* __restrict__ out) {
    int g = blockIdx.x * blockDim.x + threadIdx.x;
    if (g >= N_GRAPHS) return;
    float inv = 1.0f / fmaxf(counts[g], 1.0f);
    float acc = 0.0f;
    #pragma unroll
    for (int c = 0; c < DIM; ++c) acc += pooled[(size_t)g * DIM + c] * Wh[c];
    out[g] = acc * inv + bh[0];
}

extern "C" void kernel_launch(void* const* d_in, const int* in_sizes, int n_in,
                              void* d_out, int out_size, void* d_ws, size_t ws_size,
                              hipStream_t stream) {
    (void)in_sizes; (void)n_in; (void)out_size; (void)ws_size;
    const float* x    = (const float*)d_in[0];
    const int*   ei   = (const int*)d_in[1];
    const int*   batch= (const int*)d_in[2];
    const float* W1   = (const float*)d_in[3];
    const float* b1   = (const float*)d_in[4];
    const float* g1   = (const float*)d_in[5];
    const float* bt1  = (const float*)d_in[6];
    const float* W2   = (const float*)d_in[7];
    const float* b2   = (const float*)d_in[8];
    const float* g2   = (const float*)d_in[9];
    const float* bt2  = (const float*)d_in[10];
    const float* eps  = (const float*)d_in[11];
    const float* Wh   = (const float*)d_in[12];
    const float* bh   = (const float*)d_in[13];
    float* out = (float*)d_out;

    const int* src = ei;
    const int* dst = ei + N_EDGES;

    const size_t nfeat = (size_t)N_NODES * DIM;
    float* agg    = (float*)d_ws;
    float* z      = agg + nfeat;
    float* h      = z + nfeat;
    float* pooled = h + nfeat;
    float* counts = pooled + (size_t)N_GRAPHS * DIM;
    float* stats  = counts + N_GRAPHS;   // 256 floats
    float* colsum = stats;
    float* colsq  = stats + 64;
    float* scale  = stats + 128;
    float* shift  = stats + 192;

    const int gridFeat  = (int)((nfeat + 255) / 256);                 // 25000
    const int gridEdge  = (int)(((size_t)N_EDGES * 32 + 255) / 256);  // 200000
    const int gridTiles = N_NODES / 16;                               // 6250 (exact)

    const float* hin = x;
    for (int l = 0; l < LAYERS; ++l) {
        k_init_agg<<<gridFeat, 256, 0, stream>>>(hin, agg, eps, l);
        k_scatter<<<gridEdge, 256, 0, stream>>>(hin, src, dst, agg);

        k_zero<<<1, 256, 0, stream>>>(stats, 128);
        k_gemm_stats<<<gridTiles, 128, 0, stream>>>(agg, W1 + (size_t)l * DIM * DIM,
                                                    b1 + l * DIM, z, colsum, colsq);
        k_bn_params<<<1, 64, 0, stream>>>(colsum, colsq, g1 + l * DIM, bt1 + l * DIM,
                                          scale, shift);
        k_bnrelu<<<gridFeat, 256, 0, stream>>>(z, scale, shift, agg);

        k_zero<<<1, 256, 0, stream>>>(stats, 128);
        k_gemm_stats<<<gridTiles, 128, 0, stream>>>(agg, W2 + (size_t)l * DIM * DIM,
                                                    b2 + l * DIM, z, colsum, colsq);
        k_bn_params<<<1, 64, 0, stream>>>(colsum, colsq, g2 + l * DIM, bt2 + l * DIM,
                                          scale, shift);
        k_bnrelu<<<gridFeat, 256, 0, stream>>>(z, scale, shift, h);
        hin = h;
    }

    const int nPool = N_GRAPHS * DIM + N_GRAPHS;  // pooled + counts are contiguous
    k_zero<<<(nPool + 255) / 256, 256, 0, stream>>>(pooled, nPool);
    k_pool<<<gridFeat, 256, 0, stream>>>(h, batch, pooled, counts);
    k_head<<<(N_GRAPHS + 255) / 256, 256, 0, stream>>>(pooled, counts, Wh, bh, out);
}